// FCOS_28905129902859
// MI455X (gfx1250) — compile-verified
//
#include <hip/hip_runtime.h>
#include <stdint.h>

// ---------------- problem constants (match reference) ----------------
#define NUM_B   16
#define NUM_C   80
#define NLEV    5
#define TOPK_K  1000
#define DETS    100
#define CAND    (NLEV * TOPK_K)      // 5000 candidates per image
#define NBINS   4096
#define SCORE_T 0.2f
#define NMS_T   0.6f
#define IMG_F   1024.0f
#define OFFC    (IMG_F + 1.0f)       // class-offset trick constant

#define PIPE    4                    // async pipeline depth (LDS slots per thread)

// ---------------- CDNA5 async global->LDS path (guarded) ----------------
#if defined(__has_builtin)
#  if __has_builtin(__builtin_amdgcn_global_load_async_to_lds_b128) && \
      __has_builtin(__builtin_amdgcn_s_wait_asynccnt)
#    define USE_ASYNC_LDS 1
#  endif
#endif
#ifndef USE_ASYNC_LDS
#  define USE_ASYNC_LDS 0
#endif

#if USE_ASYNC_LDS
// Builtin signature (from compiler diagnostic): typed v4i32 pointers,
// param0 = global (AS1) source, param1 = LDS (AS3) destination, i32 offset,
// i32 cpol.
typedef int v4i_t __attribute__((ext_vector_type(4)));
typedef __attribute__((address_space(1))) v4i_t g_v4i;
typedef __attribute__((address_space(3))) v4i_t l_v4i;
#endif

static __device__ __forceinline__ void issue16(const float* g, float* lds_slot)
{
#if USE_ASYNC_LDS
    __builtin_amdgcn_global_load_async_to_lds_b128((g_v4i*)g, (l_v4i*)lds_slot,
                                                   /*offset=*/0, /*cpol=*/0);
#else
    (void)g; (void)lds_slot;
#endif
}

// s_wait_asynccnt needs a constant immediate -> template parameter
template <int N>
static __device__ __forceinline__ void wait_async_le()
{
#if USE_ASYNC_LDS
    __builtin_amdgcn_s_wait_asynccnt(N);   // async loads complete in order
#endif
}

static __device__ __forceinline__ float4 read16(const float* g, const float* lds_slot)
{
#if USE_ASYNC_LDS
    (void)g;
    return *(const float4*)lds_slot;
#else
    (void)lds_slot;
    __builtin_prefetch(g, 0, 1);           // global_prefetch_b8 (fallback path)
    return *(const float4*)g;
#endif
}

static __device__ __forceinline__ float sigmoidf_fast(float x)
{
    return 1.0f / (1.0f + __expf(-x));
}

// =====================================================================
// K0: workspace init (deterministic every call; graph-replay safe)
// =====================================================================
__global__ void k_init(uint32_t* __restrict__ hist, int2* __restrict__ cnt,
                       float* __restrict__ cscore, float* __restrict__ cbox,
                       float* __restrict__ cnb, float* __restrict__ carea,
                       float* __restrict__ clabel)
{
    const int i = blockIdx.x * blockDim.x + threadIdx.x;
    const int n = gridDim.x * blockDim.x;
    for (int k = i; k < NUM_B * NLEV * NBINS; k += n) hist[k] = 0u;
    for (int k = i; k < NUM_B * NLEV;         k += n) cnt[k] = make_int2(0, 0);
    for (int k = i; k < NUM_B * CAND;         k += n) {
        cscore[k] = -__builtin_inff();
        carea[k]  = 0.0f;
        clabel[k] = -1.0f;
    }
    for (int k = i; k < NUM_B * CAND * 4; k += n) { cbox[k] = 0.0f; cnb[k] = 0.0f; }
}

// =====================================================================
// K1: per-(image,level) score histogram (pass 1 of radix-style select)
// grid = (tiles, B), block = 256. 4-deep async-to-LDS pipeline.
// =====================================================================
__global__ void __launch_bounds__(256)
k_hist(const float* __restrict__ cls, const float* __restrict__ ctr,
       uint32_t* __restrict__ hist, int HW, int level)
{
    const int b  = blockIdx.y;
    const int N  = HW * NUM_C;
    const int NV = N >> 2;                    // float4 elements; NV % 256 == 0

    __shared__ uint32_t lh[NBINS];            // 16 KB local histogram
    __shared__ float    tile[PIPE * 256 * 4]; // 16 KB rotating async staging

    for (int i = threadIdx.x; i < NBINS; i += blockDim.x) lh[i] = 0u;
    __syncthreads();

    const float* clsb = cls + (size_t)b * N;
    const float* ctrb = ctr + (size_t)b * HW;
    const int    step = gridDim.x * blockDim.x;
    const int    v0   = blockIdx.x * blockDim.x + threadIdx.x;  // < NV (gx <= NV/256)
    float* slot[PIPE];
#pragma unroll
    for (int p = 0; p < PIPE; ++p) slot[p] = &tile[p * 1024 + threadIdx.x * 4];

    // prologue: issue PIPE-1 tiles ahead (clamped to valid addresses)
#pragma unroll
    for (int p = 0; p < PIPE - 1; ++p) {
        const int vp = v0 + p * step;
        issue16(clsb + 4 * (size_t)((vp < NV) ? vp : v0), slot[p]);
    }

    int it = 0;
    for (int v = v0; v < NV; v += step, ++it) {
        // issue tile v + (PIPE-1)*step into the slot consumed last iteration
        const int vn = v + (PIPE - 1) * step;
        issue16(clsb + 4 * (size_t)((vn < NV) ? vn : v), slot[(it + PIPE - 1) % PIPE]);
        wait_async_le<PIPE - 1>();            // oldest tile (v) has landed
        const float4 d = read16(clsb + 4 * (size_t)v, slot[it % PIPE]);

        const int idx0 = 4 * v;
        const float dv[4] = {d.x, d.y, d.z, d.w};
#pragma unroll
        for (int k = 0; k < 4; ++k) {
            const int   idx = idx0 + k;
            const int   a   = idx / NUM_C;
            const float s   = __fsqrt_rn(sigmoidf_fast(dv[k]) * sigmoidf_fast(ctrb[a]));
            if (s > SCORE_T) {
                int bin = (int)(s * (float)NBINS);
                bin = bin > (NBINS - 1) ? (NBINS - 1) : bin;
                atomicAdd(&lh[bin], 1u);
            }
        }
    }
    wait_async_le<0>();                       // drain before LDS reuse
    __syncthreads();

    uint32_t* gh = hist + (size_t)(b * NLEV + level) * NBINS;
    for (int i = threadIdx.x; i < NBINS; i += blockDim.x)
        if (lh[i]) atomicAdd(&gh[i], lh[i]);
}

// =====================================================================
// K2: find per-(image,level) cutoff bucket + rank-within-bucket
// =====================================================================
__global__ void k_cut(const uint32_t* __restrict__ hist, int2* __restrict__ cut)
{
    const int t = blockIdx.x * blockDim.x + threadIdx.x;
    if (t >= NUM_B * NLEV) return;
    const uint32_t* h  = hist + (size_t)t * NBINS;
    const int       lo = (int)(SCORE_T * (float)NBINS);   // 819: lowest bin > thresh
    int cum = 0, tb = lo - 1, r = 0;                      // default: accept all bins >= lo
    for (int i = NBINS - 1; i >= lo; --i) {
        const int c = (int)h[i];
        if (cum + c >= TOPK_K) { tb = i; r = TOPK_K - cum; break; }
        cum += c;
    }
    cut[t] = make_int2(tb, r);
}

// =====================================================================
// K3: pass 2 — collect top-K candidates, decode + clip boxes, prepare
// class-offset boxes and areas for NMS. Same 4-deep async pipeline.
// =====================================================================
__global__ void __launch_bounds__(256)
k_collect(const float* __restrict__ cls, const float* __restrict__ ctr,
          const float* __restrict__ reg, const float* __restrict__ anc,
          const int2* __restrict__ cut, int2* __restrict__ cnt,
          float* __restrict__ cscore, float* __restrict__ cbox,
          float* __restrict__ cnb, float* __restrict__ carea,
          float* __restrict__ clabel, int HW, int level)
{
    const int b  = blockIdx.y;
    const int N  = HW * NUM_C;
    const int NV = N >> 2;
    const int bl = b * NLEV + level;

    __shared__ float tile[PIPE * 256 * 4];

    const float* clsb = cls + (size_t)b * N;
    const float* ctrb = ctr + (size_t)b * HW;
    const int2   ct   = cut[bl];
    const int    step = gridDim.x * blockDim.x;
    const int    v0   = blockIdx.x * blockDim.x + threadIdx.x;
    float* slot[PIPE];
#pragma unroll
    for (int p = 0; p < PIPE; ++p) slot[p] = &tile[p * 1024 + threadIdx.x * 4];

#pragma unroll
    for (int p = 0; p < PIPE - 1; ++p) {
        const int vp = v0 + p * step;
        issue16(clsb + 4 * (size_t)((vp < NV) ? vp : v0), slot[p]);
    }

    int it = 0;
    for (int v = v0; v < NV; v += step, ++it) {
        const int vn = v + (PIPE - 1) * step;
        issue16(clsb + 4 * (size_t)((vn < NV) ? vn : v), slot[(it + PIPE - 1) % PIPE]);
        wait_async_le<PIPE - 1>();
        const float4 d = read16(clsb + 4 * (size_t)v, slot[it % PIPE]);

        const int idx0 = 4 * v;
        const float dv[4] = {d.x, d.y, d.z, d.w};
#pragma unroll
        for (int k = 0; k < 4; ++k) {
            const int   idx = idx0 + k;
            const int   a   = idx / NUM_C;
            const float s   = __fsqrt_rn(sigmoidf_fast(dv[k]) * sigmoidf_fast(ctrb[a]));
            if (!(s > SCORE_T)) continue;
            int bin = (int)(s * (float)NBINS);
            bin = bin > (NBINS - 1) ? (NBINS - 1) : bin;

            int pos = -1;
            if (bin > ct.x) {
                pos = atomicAdd(&cnt[bl].x, 1);
            } else if (bin == ct.x) {
                const int e = atomicAdd(&cnt[bl].y, 1);
                if (e < ct.y) pos = atomicAdd(&cnt[bl].x, 1);
            }
            if (pos < 0 || pos >= TOPK_K) continue;

            const int ccls = idx - a * NUM_C;             // idx % C
            // decode (BoxLinearCoder, normalize_by_size=True)
            const float4 A  = ((const float4*)anc)[a];
            const float  cx = (A.x + A.z) * 0.5f, cy = (A.y + A.w) * 0.5f;
            const float  w  = A.z - A.x,          h  = A.w - A.y;
            const float4 R  = ((const float4*)reg)[(size_t)b * HW + a];
            float x1 = cx - R.x * w, y1 = cy - R.y * h;
            float x2 = cx + R.z * w, y2 = cy + R.w * h;
            x1 = fminf(fmaxf(x1, 0.0f), IMG_F); y1 = fminf(fmaxf(y1, 0.0f), IMG_F);
            x2 = fminf(fmaxf(x2, 0.0f), IMG_F); y2 = fminf(fmaxf(y2, 0.0f), IMG_F);

            const size_t slotw = (size_t)b * CAND + (size_t)level * TOPK_K + pos;
            const float  off   = OFFC * (float)ccls;
            ((float4*)cbox)[slotw] = make_float4(x1, y1, x2, y2);
            ((float4*)cnb)[slotw]  = make_float4(x1 + off, y1 + off, x2 + off, y2 + off);
            carea[slotw]  = (x2 - x1) * (y2 - y1);
            cscore[slotw] = s;
            clabel[slotw] = (float)ccls;
        }
    }
    wait_async_le<0>();
}

// =====================================================================
// K4: per-image greedy class-batched NMS (scores live in LDS)
// grid = B, block = 256
// =====================================================================
__global__ void __launch_bounds__(256)
k_nms(const float* __restrict__ cscore, const float* __restrict__ cbox,
      const float* __restrict__ cnb, const float* __restrict__ carea,
      const float* __restrict__ clabel, float* __restrict__ out)
{
    const int b = blockIdx.x;
    __shared__ float s[CAND];         // 20 KB (CDNA5 WGP has 320 KB)
    __shared__ float rv[256];
    __shared__ int   ri[256];
    __shared__ int   sj;
    __shared__ float sjs;

    for (int i = threadIdx.x; i < CAND; i += 256) s[i] = cscore[(size_t)b * CAND + i];
    __syncthreads();

    const float4* nb = (const float4*)cnb + (size_t)b * CAND;
    const float*  ar = carea + (size_t)b * CAND;
    float* ob = out;                              // [B][DETS][4]
    float* os = out + (size_t)NUM_B * DETS * 4;   // [B][DETS]
    float* ol = out + (size_t)NUM_B * DETS * 5;   // [B][DETS]

    for (int d = 0; d < DETS; ++d) {
        // parallel argmax over CAND scores
        float bv = -__builtin_inff(); int bi = 0;
        for (int i = threadIdx.x; i < CAND; i += 256) {
            const float v = s[i];
            if (v > bv) { bv = v; bi = i; }
        }
        rv[threadIdx.x] = bv; ri[threadIdx.x] = bi;
        __syncthreads();
        for (int off = 128; off > 0; off >>= 1) {
            if (threadIdx.x < off && rv[threadIdx.x + off] > rv[threadIdx.x]) {
                rv[threadIdx.x] = rv[threadIdx.x + off];
                ri[threadIdx.x] = ri[threadIdx.x + off];
            }
            __syncthreads();
        }
        if (threadIdx.x == 0) { sj = ri[0]; sjs = rv[0]; }
        __syncthreads();
        const int   j  = sj;
        const float js = sjs;

        if (threadIdx.x == 0) {
            const bool   valid = js > SCORE_T;
            const float4 B4    = ((const float4*)cbox)[(size_t)b * CAND + j];
            float* o = ob + ((size_t)b * DETS + d) * 4;
            o[0] = valid ? B4.x : 0.0f;  o[1] = valid ? B4.y : 0.0f;
            o[2] = valid ? B4.z : 0.0f;  o[3] = valid ? B4.w : 0.0f;
            os[(size_t)b * DETS + d] = valid ? js : 0.0f;
            ol[(size_t)b * DETS + d] = valid ? clabel[(size_t)b * CAND + j] : -1.0f;
        }

        // suppress everything with IoU > thresh vs j (including j itself)
        const float4 J  = nb[j];
        const float  aj = ar[j];
        for (int i = threadIdx.x; i < CAND; i += 256) {
            const float4 Ni  = nb[i];
            const float xx1 = fmaxf(J.x, Ni.x), yy1 = fmaxf(J.y, Ni.y);
            const float xx2 = fminf(J.z, Ni.z), yy2 = fminf(J.w, Ni.w);
            const float inter = fmaxf(xx2 - xx1, 0.0f) * fmaxf(yy2 - yy1, 0.0f);
            const float iou   = inter / (ar[i] + aj - inter + 1e-9f);
            if (iou > NMS_T) s[i] = -__builtin_inff();
        }
        __syncthreads();
    }
}

// =====================================================================
// host launcher
// =====================================================================
extern "C" void kernel_launch(void* const* d_in, const int* in_sizes, int n_in,
                              void* d_out, int out_size, void* d_ws, size_t ws_size,
                              hipStream_t stream)
{
    (void)in_sizes; (void)n_in; (void)out_size; (void)ws_size;
    static const int HWs[NLEV] = {16384, 4096, 1024, 256, 64};

    // ---- workspace layout (256B aligned slabs) ----
    uint8_t* w = (uint8_t*)d_ws;
    size_t off = 0;
    auto take = [&](size_t bytes) -> void* {
        void* p = w + off;
        off += (bytes + 255) & ~(size_t)255;
        return p;
    };
    uint32_t* hist   = (uint32_t*)take((size_t)NUM_B * NLEV * NBINS * sizeof(uint32_t));
    int2*     cut    = (int2*)    take((size_t)NUM_B * NLEV * sizeof(int2));
    int2*     cnt    = (int2*)    take((size_t)NUM_B * NLEV * sizeof(int2));
    float*    cscore = (float*)   take((size_t)NUM_B * CAND * sizeof(float));
    float*    cbox   = (float*)   take((size_t)NUM_B * CAND * 4 * sizeof(float));
    float*    cnb    = (float*)   take((size_t)NUM_B * CAND * 4 * sizeof(float));
    float*    carea  = (float*)   take((size_t)NUM_B * CAND * sizeof(float));
    float*    clabel = (float*)   take((size_t)NUM_B * CAND * sizeof(float));

    k_init<<<256, 256, 0, stream>>>(hist, cnt, cscore, cbox, cnb, carea, clabel);

    // pass 1: histograms
    for (int l = 0; l < NLEV; ++l) {
        const int HW = HWs[l];
        const int NV = (HW * NUM_C) >> 2;
        int gx = NV / 256; if (gx > 240) gx = 240; if (gx < 1) gx = 1;
        const float* cls = (const float*)d_in[4 * l + 0];
        const float* ctr = (const float*)d_in[4 * l + 1];
        k_hist<<<dim3(gx, NUM_B), 256, 0, stream>>>(cls, ctr, hist, HW, l);
    }

    k_cut<<<(NUM_B * NLEV + 127) / 128, 128, 0, stream>>>(hist, cut);

    // pass 2: collect + decode
    for (int l = 0; l < NLEV; ++l) {
        const int HW = HWs[l];
        const int NV = (HW * NUM_C) >> 2;
        int gx = NV / 256; if (gx > 240) gx = 240; if (gx < 1) gx = 1;
        const float* cls = (const float*)d_in[4 * l + 0];
        const float* ctr = (const float*)d_in[4 * l + 1];
        const float* reg = (const float*)d_in[4 * l + 2];
        const float* anc = (const float*)d_in[4 * l + 3];
        k_collect<<<dim3(gx, NUM_B), 256, 0, stream>>>(cls, ctr, reg, anc, cut, cnt,
                                                       cscore, cbox, cnb, carea, clabel,
                                                       HW, l);
    }

    k_nms<<<NUM_B, 256, 0, stream>>>(cscore, cbox, cnb, carea, clabel, (float*)d_out);
}